// CondensationLossTiger_37383395344509
// MI455X (gfx1250) — compile-verified
//
#include <hip/hip_runtime.h>

typedef float v2f __attribute__((ext_vector_type(2)));
typedef float v8f __attribute__((ext_vector_type(8)));

#define NCLUST 1024           // K (cluster id 0 = noise)
#define NOBJ   1023           // K-1 objects
#define NCOLP  1024           // objects padded to multiple of 16
#define DEMB   32
#define NGRP   8              // column groups per row tile (8 tiles each)
#define QMIN   0.01f
#define PTTHLD 0.9f
#define MAXETA 4.0f
#define EPSF   1e-9f

// ---------------- kernel 1: init per-object argmax keys -------------------
// key = (float_bits(q) << 32) | (0xFFFFFFFF - j): larger q wins, ties -> smaller j.
// init = q bits 0, j = 0 -> empty object decodes to alpha = 0 (matches jnp.argmax of zeros).
__global__ void ct_initkeys(unsigned long long* __restrict__ keys) {
    int c = blockIdx.x * blockDim.x + threadIdx.x;
    if (c < NOBJ) keys[c] = 0x00000000FFFFFFFFull;
}

// ---------------- kernel 2: per-hit prep ----------------------------------
__global__ void ct_prep(const float* __restrict__ beta, const float* __restrict__ x,
                        const float* __restrict__ pt, const float* __restrict__ eta,
                        const int* __restrict__ recon, const int* __restrict__ cids,
                        float* __restrict__ qhit, float* __restrict__ xnorm,
                        int* __restrict__ amatch, unsigned long long* __restrict__ keys,
                        int N) {
    int j = blockIdx.x * blockDim.x + threadIdx.x;
    if (j >= N) return;
    float braw = beta[j];
    float bc = fminf(fmaxf(braw, 0.0f), 1.0f - 1e-4f);
    float at = atanhf(bc);
    float q = fmaf(at, at, QMIN);
    bool ok = (recon[j] > 0) && (pt[j] > PTTHLD) && (fabsf(eta[j]) < MAXETA);
    int cid = cids[j];
    const float* xr = x + (size_t)j * DEMB;
    float s = 0.0f;
#pragma unroll
    for (int d = 0; d < DEMB; ++d) s = fmaf(xr[d], xr[d], s);
    qhit[j] = q;
    xnorm[j] = s;
    amatch[j] = (ok && cid > 0) ? cid : 0;   // 0 matches no object id (ids are 1..1023)
    if (ok && cid >= 1 && cid <= NOBJ) {
        unsigned long long key =
            ((unsigned long long)__float_as_uint(q) << 32) |
            (unsigned long long)(0xFFFFFFFFu - (unsigned)j);
        atomicMax(&keys[cid - 1], key);      // integer max: deterministic
    }
}

// ---------------- kernel 3: gather condensation points --------------------
__global__ void ct_gather(const float* __restrict__ x, const float* __restrict__ qhit,
                          const float* __restrict__ xnorm,
                          const unsigned long long* __restrict__ keys,
                          float* __restrict__ qk, float* __restrict__ knorm,
                          float* __restrict__ Xk, int* __restrict__ alphas) {
    int c = blockIdx.x * blockDim.x + threadIdx.x;
    if (c >= NCOLP) return;
    float* dst = Xk + (size_t)c * DEMB;
    if (c < NOBJ) {
        unsigned long long key = keys[c];
        unsigned alpha = 0xFFFFFFFFu - (unsigned)(key & 0xFFFFFFFFull);
        alphas[c] = (int)alpha;
        qk[c] = qhit[alpha];       // q of the condensation hit (q[0] for empty objects)
        knorm[c] = xnorm[alpha];
        const float* src = x + (size_t)alpha * DEMB;
#pragma unroll
        for (int d = 0; d < DEMB; ++d) dst[d] = src[d];
    } else {                        // pad column: q_k = 0 -> contributes 0 everywhere
        qk[c] = 0.0f;
        knorm[c] = 0.0f;
#pragma unroll
        for (int d = 0; d < DEMB; ++d) dst[d] = 0.0f;
    }
}

// ---------------- kernel 4: tiled WMMA Gram matrix + fused epilogue -------
// One wave owns a 16x16 tile of S = x_tile (16x32) @ Xk_tile^T (32x16),
// built from two independent chains of 4x V_WMMA_F32_16X16X4_F32 (ILP).
// Each wave sweeps 8 column tiles so A regs + row metadata load once.
__global__ void __launch_bounds__(256)
ct_main(const float* __restrict__ x, const float* __restrict__ qhit,
        const float* __restrict__ xnorm, const int* __restrict__ amatch,
        const float* __restrict__ Xk, const float* __restrict__ qk,
        const float* __restrict__ knorm, float* __restrict__ partials,
        int nRowTiles, int N) {
    int wave = (blockIdx.x * blockDim.x + threadIdx.x) >> 5;
    int lane = threadIdx.x & 31;
    int nWaves = nRowTiles * NGRP;
    if (wave >= nWaves) return;                 // whole-wave uniform exit
    int rtile  = wave / NGRP;
    int cgroup = wave % NGRP;
    int half = lane >> 4;                       // K-half / row-half selector
    int r0   = lane & 15;

    // A: lane holds row (rtile*16 + r0); per 4-wide K block, VGPR pair holds
    // K = {2*half, 2*half+1} (ISA 7.12.2, 32-bit A-matrix layout).
    int arow_i = rtile * 16 + r0;
    v2f a[8];
#pragma unroll
    for (int t = 0; t < 8; ++t) {
        if (arow_i < N)
            a[t] = *(const v2f*)(x + (size_t)arow_i * DEMB + t * 4 + half * 2);
        else
            a[t] = (v2f){0.0f, 0.0f};           // padded rows: q=0 below -> zero contrib
    }

    // Row metadata per the C/D layout: VGPR v holds row v + 8*half.
    float qrow[8], xn[8];
    int am[8];
#pragma unroll
    for (int v = 0; v < 8; ++v) {
        int row = rtile * 16 + v + 8 * half;
        if (row < N) { qrow[v] = qhit[row]; xn[v] = xnorm[row]; am[v] = amatch[row]; }
        else         { qrow[v] = 0.0f;      xn[v] = 0.0f;       am[v] = -1; }
    }

    float vatt = 0.0f, vrep = 0.0f;
    const int ct0 = cgroup * (64 / NGRP);
    for (int ct = ct0; ct < ct0 + (64 / NGRP); ++ct) {
        int col = ct * 16 + r0;                 // lane's column (C/D layout: col = lane%16)
        const float* brow = Xk + (size_t)col * DEMB;
        v2f b[8];
#pragma unroll
        for (int t = 0; t < 8; ++t)
            b[t] = *(const v2f*)(brow + t * 4 + half * 2);
        float qc = qk[col], kn = knorm[col];
        int objid = col + 1;                    // object ids 1..1023; pad col never matches

        // two independent WMMA accumulator chains for matrix-pipe ILP
        v8f c0 = {}, c1 = {};
#pragma unroll
        for (int t = 0; t < 4; ++t) {
            c0 = __builtin_amdgcn_wmma_f32_16x16x4_f32(
                     false, a[2 * t],     false, b[2 * t],     (short)0, c0, false, false);
            c1 = __builtin_amdgcn_wmma_f32_16x16x4_f32(
                     false, a[2 * t + 1], false, b[2 * t + 1], (short)0, c1, false, false);
        }

#pragma unroll
        for (int v = 0; v < 8; ++v) {
            float s    = c0[v] + c1[v];
            float d2   = xn[v] + kn - 2.0f * s;
            float d2c  = fmaxf(d2, 1e-12f);
            float dist = __builtin_amdgcn_sqrtf(d2c);      // raw v_sqrt_f32
            float qw   = qrow[v] * qc;
            float hinge = fmaxf(1.0f - dist, 0.0f);        // == (dist<1)*(1-dist)
            bool att   = (am[v] == objid);
            float ca   = att ? qw * d2c : 0.0f;            // v_cndmask, no branch
            float cr   = att ? 0.0f : qw * hinge;          // v_cndmask, no branch
            vatt += ca;
            vrep += cr;
        }
    }

    // wave32 shuffle reduction, one partial pair per wave (deterministic)
#pragma unroll
    for (int off = 16; off; off >>= 1) {
        vatt += __shfl_down(vatt, off, 32);
        vrep += __shfl_down(vrep, off, 32);
    }
    if (lane == 0) {
        partials[2 * wave]     = vatt;
        partials[2 * wave + 1] = vrep;
    }
}

// ---------------- kernel 5: deterministic finalize ------------------------
__global__ void __launch_bounds__(256)
ct_final(const float* __restrict__ beta, const float* __restrict__ pt,
         const float* __restrict__ eta, const int* __restrict__ recon,
         const int* __restrict__ cids, const float* __restrict__ partials,
         int nWaves, const int* __restrict__ alphas, float* __restrict__ out, int N) {
    __shared__ float s_att[256], s_rep[256], s_noise[256], s_cow[256];
    __shared__ int   s_ncnt[256], s_ok[256];
    int t = threadIdx.x;
    float att = 0.0f, rep = 0.0f, noise = 0.0f, cow = 0.0f;
    int ncnt = 0, okc = 0;
    for (int w = t; w < nWaves; w += 256) { att += partials[2 * w]; rep += partials[2 * w + 1]; }
    for (int j = t; j < N; j += 256) {
        float b = beta[j];
        if (cids[j] <= 0) { noise += b; ncnt++; }
        if ((recon[j] > 0) && (pt[j] > PTTHLD) && (fabsf(eta[j]) < MAXETA)) okc++;
    }
    for (int c = t; c < NOBJ; c += 256) cow += 1.0f - beta[alphas[c]];
    s_att[t] = att; s_rep[t] = rep; s_noise[t] = noise; s_cow[t] = cow;
    s_ncnt[t] = ncnt; s_ok[t] = okc;
    __syncthreads();
    for (int off = 128; off; off >>= 1) {
        if (t < off) {
            s_att[t] += s_att[t + off]; s_rep[t] += s_rep[t + off];
            s_noise[t] += s_noise[t + off]; s_cow[t] += s_cow[t + off];
            s_ncnt[t] += s_ncnt[t + off]; s_ok[t] += s_ok[t + off];
        }
        __syncthreads();
    }
    if (t == 0) {
        float n_part   = (float)NOBJ;
        float norm_att = EPSF + (float)s_ok[0] - n_part;
        float norm_rep = EPSF + (n_part - 1.0f) * (float)N;
        float l_cow    = s_cow[0] / n_part;
        float l_noise  = s_noise[0] / fmaxf((float)s_ncnt[0], 1.0f);
        out[0] = s_att[0] / norm_att + 1.0f * (s_rep[0] / norm_rep)
               + 0.1f * l_noise + 0.1f * l_cow;
    }
}

// ---------------- launcher ------------------------------------------------
extern "C" void kernel_launch(void* const* d_in, const int* in_sizes, int n_in,
                              void* d_out, int out_size, void* d_ws, size_t ws_size,
                              hipStream_t stream) {
    const float* beta  = (const float*)d_in[0];
    const float* x     = (const float*)d_in[1];
    const float* pt    = (const float*)d_in[2];
    const float* eta   = (const float*)d_in[3];
    const int*   recon = (const int*)d_in[4];
    const int*   cids  = (const int*)d_in[5];
    const int N = in_sizes[0];
    const int nRowTiles = (N + 15) / 16;
    const int nWaves = nRowTiles * NGRP;

    // workspace carve-out (256B aligned slices)
    char* ws = (char*)d_ws;
    size_t off = 0;
    auto take = [&](size_t bytes) -> char* {
        char* p = ws + off;
        off = (off + bytes + 255) & ~(size_t)255;
        return p;
    };
    unsigned long long* keys = (unsigned long long*)take(NOBJ * 8);
    float* qk     = (float*)take(NCOLP * 4);
    float* knorm  = (float*)take(NCOLP * 4);
    float* Xk     = (float*)take((size_t)NCOLP * DEMB * 4);
    int*   alphas = (int*)take(NOBJ * 4);
    float* qhit   = (float*)take((size_t)N * 4);
    float* xnorm  = (float*)take((size_t)N * 4);
    int*   amatch = (int*)take((size_t)N * 4);
    float* partials = (float*)take((size_t)nWaves * 2 * 4);
    (void)ws_size; (void)n_in; (void)out_size;

    ct_initkeys<<<(NOBJ + 255) / 256, 256, 0, stream>>>(keys);
    ct_prep<<<(N + 255) / 256, 256, 0, stream>>>(beta, x, pt, eta, recon, cids,
                                                 qhit, xnorm, amatch, keys, N);
    ct_gather<<<NCOLP / 256, 256, 0, stream>>>(x, qhit, xnorm, keys, qk, knorm, Xk, alphas);
    ct_main<<<(nWaves * 32 + 255) / 256, 256, 0, stream>>>(x, qhit, xnorm, amatch,
                                                           Xk, qk, knorm, partials,
                                                           nRowTiles, N);
    ct_final<<<1, 256, 0, stream>>>(beta, pt, eta, recon, cids, partials, nWaves,
                                    alphas, (float*)d_out, N);
}